// LATTICE_42932493091123
// MI455X (gfx1250) — compile-verified
//
#include <hip/hip_runtime.h>
#include <hip/hip_bf16.h>

#define NUM_USER 20000
#define NUM_ITEM 6144
#define DIM_E 64
#define TOPK 10
#define LAMBDA 0.9f
#define V_DIM 4096
#define T_DIM 384
#define N_EDGE 1000000
#define NUM_NODES (NUM_USER + NUM_ITEM)

typedef __attribute__((ext_vector_type(2))) float v2f;
typedef __attribute__((ext_vector_type(8))) float v8f;

__device__ __forceinline__ v8f wmma_f32_4(v2f a, v2f b, v8f c) {
    // D = A(16x4,f32) * B(4x16,f32) + C(16x16,f32)
    return __builtin_amdgcn_wmma_f32_16x16x4_f32(false, a, false, b, (short)0, c,
                                                 false, false);
}

// --- CDNA5 async global->LDS copy (ASYNCcnt-tracked), inline asm -----------
__device__ __forceinline__ void async_copy_b128(unsigned lds_off, const float* gaddr) {
    asm volatile("global_load_async_to_lds_b128 %0, %1, off"
                 :: "v"(lds_off), "v"(gaddr)
                 : "memory");
}
__device__ __forceinline__ void wait_async_le16() {
    asm volatile("s_wait_asynccnt 0x10" ::: "memory");
}
__device__ __forceinline__ void wait_async_0() {
    asm volatile("s_wait_asynccnt 0x0" ::: "memory");
}

// ---------------------------------------------------------------------------
// K1: feats = X(M x K) @ W(K x 64) + bias     (one wave / 16-row tile)
// ---------------------------------------------------------------------------
__global__ void lattice_proj_kernel(const float* __restrict__ X,
                                    const float* __restrict__ W,
                                    const float* __restrict__ bias,
                                    float* __restrict__ out, int K) {
    const int row0 = blockIdx.x * 16;
    const int lane = threadIdx.x;
    const int half = lane >> 4;
    const int l15  = lane & 15;
    v8f acc[4] = {};
    for (int k = 0; k < K; k += 4) {
        v2f a;
        const float* ap = X + (size_t)(row0 + l15) * K + k + half * 2;
        a.x = ap[0];
        a.y = ap[1];
#pragma unroll
        for (int nt = 0; nt < 4; ++nt) {
            const int n = nt * 16 + l15;
            v2f b;
            b.x = W[(size_t)(k + half * 2) * DIM_E + n];
            b.y = W[(size_t)(k + half * 2 + 1) * DIM_E + n];
            acc[nt] = wmma_f32_4(a, b, acc[nt]);
        }
    }
#pragma unroll
    for (int nt = 0; nt < 4; ++nt) {
        const int n = nt * 16 + l15;
#pragma unroll
        for (int r = 0; r < 8; ++r) {
            const int m = row0 + r + half * 8;
            out[(size_t)m * DIM_E + n] = acc[nt][r] + bias[n];
        }
    }
}

// ---------------------------------------------------------------------------
// K2: row-wise L2 normalize (one wave / row, 64 dims)
// ---------------------------------------------------------------------------
__global__ void lattice_rownorm_kernel(const float* __restrict__ in,
                                       float* __restrict__ out) {
    const int row = blockIdx.x;
    const int lane = threadIdx.x;
    const size_t o = (size_t)row * DIM_E;
    float x0 = in[o + lane];
    float x1 = in[o + 32 + lane];
    float ss = x0 * x0 + x1 * x1;
#pragma unroll
    for (int off = 16; off > 0; off >>= 1) ss += __shfl_xor(ss, off, 32);
    const float r = rsqrtf(ss);
    out[o + lane] = x0 * r;
    out[o + 32 + lane] = x1 * r;
}

// ---------------------------------------------------------------------------
// K3: fused S = Xn Xn^T + per-row top-10 (one wave / 16 rows)
// ---------------------------------------------------------------------------
__global__ void lattice_simtopk_kernel(const float* __restrict__ Xn,
                                       float* __restrict__ tv,
                                       int* __restrict__ ti) {
    __shared__ float tile[16][17];
    const int row0 = blockIdx.x * 16;
    const int lane = threadIdx.x;
    const int half = lane >> 4;
    const int l15  = lane & 15;

    v2f afrag[16];
#pragma unroll
    for (int kk = 0; kk < 16; ++kk) {
        const float* ap = Xn + (size_t)(row0 + l15) * DIM_E + kk * 4 + half * 2;
        afrag[kk].x = ap[0];
        afrag[kk].y = ap[1];
    }

    float bestv[TOPK];
    int   besti[TOPK];
#pragma unroll
    for (int t = 0; t < TOPK; ++t) { bestv[t] = -1e30f; besti[t] = 0; }

    for (int j0 = 0; j0 < NUM_ITEM; j0 += 16) {
        v8f acc = {};
#pragma unroll
        for (int kk = 0; kk < 16; ++kk) {
            const float* bp = Xn + (size_t)(j0 + l15) * DIM_E + kk * 4 + half * 2;
            v2f b;
            b.x = bp[0];
            b.y = bp[1];
            acc = wmma_f32_4(afrag[kk], b, acc);
        }
#pragma unroll
        for (int r = 0; r < 8; ++r) tile[r + half * 8][l15] = acc[r];
        __syncthreads();
        if (lane < 16) {
#pragma unroll
            for (int n = 0; n < 16; ++n) {
                const float v = tile[lane][n];
                if (v > bestv[TOPK - 1]) {
                    bestv[TOPK - 1] = v;
                    besti[TOPK - 1] = j0 + n;
#pragma unroll
                    for (int q = TOPK - 1; q > 0; --q) {
                        if (bestv[q] > bestv[q - 1]) {
                            float fv = bestv[q]; bestv[q] = bestv[q - 1]; bestv[q - 1] = fv;
                            int   iv = besti[q]; besti[q] = besti[q - 1]; besti[q - 1] = iv;
                        }
                    }
                }
            }
        }
        __syncthreads();
    }
    if (lane < 16) {
#pragma unroll
        for (int t = 0; t < TOPK; ++t) {
            tv[(size_t)(row0 + lane) * TOPK + t] = bestv[t];
            ti[(size_t)(row0 + lane) * TOPK + t] = besti[t];
        }
    }
}

// ---------------------------------------------------------------------------
// K4: learned-adj degree^(-1/2):  rowsum = w0*sum(topk_img)+w1*sum(topk_txt)
// ---------------------------------------------------------------------------
__global__ void lattice_learned_d_kernel(const float* __restrict__ tvi,
                                         const float* __restrict__ tvt,
                                         const float* __restrict__ mw,
                                         float* __restrict__ dvec) {
    const int i = blockIdx.x * blockDim.x + threadIdx.x;
    if (i >= NUM_ITEM) return;
    const float e0 = __expf(mw[0]), e1 = __expf(mw[1]);
    const float w0 = e0 / (e0 + e1), w1 = e1 / (e0 + e1);
    float s = 0.0f;
#pragma unroll
    for (int t = 0; t < TOPK; ++t)
        s += w0 * tvi[(size_t)i * TOPK + t] + w1 * tvt[(size_t)i * TOPK + t];
    dvec[i] = (s > 0.0f) ? rsqrtf(s) : 0.0f;
}

// ---------------------------------------------------------------------------
// K5a: h = (1-lambda) * learned_adj_norm @ item_emb   (sparse, <=20 nnz/row)
// ---------------------------------------------------------------------------
__global__ void lattice_h_sparse_kernel(const float* __restrict__ tvi,
                                        const int* __restrict__ tii,
                                        const float* __restrict__ tvt,
                                        const int* __restrict__ tit,
                                        const float* __restrict__ dvec,
                                        const float* __restrict__ mw,
                                        const float* __restrict__ item_emb,
                                        float* __restrict__ h) {
    const int row = blockIdx.x;
    const int lane = threadIdx.x;
    const float e0 = __expf(mw[0]), e1 = __expf(mw[1]);
    const float w0 = e0 / (e0 + e1), w1 = e1 / (e0 + e1);
    const float di = dvec[row] * (1.0f - LAMBDA);
    float a0 = 0.0f, a1 = 0.0f;
#pragma unroll
    for (int t = 0; t < TOPK; ++t) {
        const int   j  = tii[(size_t)row * TOPK + t];
        const float c  = w0 * tvi[(size_t)row * TOPK + t] * di * dvec[j];
        a0 += c * item_emb[(size_t)j * DIM_E + lane];
        a1 += c * item_emb[(size_t)j * DIM_E + 32 + lane];
        const int   j2 = tit[(size_t)row * TOPK + t];
        const float c2 = w1 * tvt[(size_t)row * TOPK + t] * di * dvec[j2];
        a0 += c2 * item_emb[(size_t)j2 * DIM_E + lane];
        a1 += c2 * item_emb[(size_t)j2 * DIM_E + 32 + lane];
    }
    h[(size_t)row * DIM_E + lane] = a0;
    h[(size_t)row * DIM_E + 32 + lane] = a1;
}

// ---------------------------------------------------------------------------
// K5b: h += lambda * (w0*A_img + w1*A_txt) @ item_emb
//      Dense 6144x6144x64 GEMM. The A stream (302 MB total) is staged
//      global->LDS with async b128 copies, double-buffered, ASYNCcnt waits.
// ---------------------------------------------------------------------------
#define HD_CHUNK 64   // K elements staged per buffer
__global__ void lattice_h_dense_kernel(const float* __restrict__ Aimg,
                                       const float* __restrict__ Atxt,
                                       const float* __restrict__ mw,
                                       const float* __restrict__ item_emb,
                                       float* __restrict__ h) {
    // abuf[buf][mat][row 0..15][k 0..63]  -> 2*2*16*64*4B = 16 KB
    __shared__ float abuf[2][2][16][HD_CHUNK];
    const int row0 = blockIdx.x * 16;
    const int lane = threadIdx.x;
    const int half = lane >> 4;
    const int l15  = lane & 15;
    const float e0 = __expf(mw[0]), e1 = __expf(mw[1]);
    const float w0 = LAMBDA * e0 / (e0 + e1), w1 = LAMBDA * e1 / (e0 + e1);

    const unsigned lds_base = (unsigned)(uintptr_t)(&abuf[0][0][0][0]);
    const unsigned buf_bytes = 2u * 16u * HD_CHUNK * 4u;

    v8f acc[4];
#pragma unroll
    for (int nt = 0; nt < 4; ++nt) {
        const int n = nt * 16 + l15;
#pragma unroll
        for (int r = 0; r < 8; ++r)
            acc[nt][r] = h[(size_t)(row0 + r + half * 8) * DIM_E + n];
    }

    // stage one chunk (16 rows x 64 k of both adjacencies) into buffer `buf`
    // 256 16-byte units per matrix -> 8 per lane per matrix
    auto stage = [&](int k0, int buf) {
        const unsigned base = lds_base + (unsigned)buf * buf_bytes;
#pragma unroll
        for (int u = 0; u < 8; ++u) {
            const int unit = u * 32 + lane;     // 0..255
            const int r    = unit >> 4;         // row 0..15
            const int koff = (unit & 15) * 4;   // k offset 0..60
            const size_t gi = (size_t)(row0 + r) * NUM_ITEM + k0 + koff;
            const unsigned li = base + (unsigned)((r * HD_CHUNK + koff) * 4);
            async_copy_b128(li, Aimg + gi);
            async_copy_b128(li + (unsigned)(16 * HD_CHUNK * 4), Atxt + gi);
        }
    };

    const int nchunks = NUM_ITEM / HD_CHUNK;   // 96
    stage(0, 0);
    for (int c = 0; c < nchunks; ++c) {
        const int buf = c & 1;
        if (c + 1 < nchunks) {
            stage((c + 1) * HD_CHUNK, 1 - buf);
            wait_async_le16();   // chunk c landed; chunk c+1 (16 ops) in flight
        } else {
            wait_async_0();
        }
        const int kbase = c * HD_CHUNK;
#pragma unroll
        for (int kk = 0; kk < HD_CHUNK / 4; ++kk) {
            const int ko = kk * 4 + half * 2;
            v2f ai = *(const v2f*)&abuf[buf][0][l15][ko];
            v2f at = *(const v2f*)&abuf[buf][1][l15][ko];
            v2f a;
            a.x = w0 * ai.x + w1 * at.x;
            a.y = w0 * ai.y + w1 * at.y;
            const int k = kbase + kk * 4;
#pragma unroll
            for (int nt = 0; nt < 4; ++nt) {
                const int n = nt * 16 + l15;
                v2f b;
                b.x = item_emb[(size_t)(k + half * 2) * DIM_E + n];
                b.y = item_emb[(size_t)(k + half * 2 + 1) * DIM_E + n];
                acc[nt] = wmma_f32_4(a, b, acc[nt]);
            }
        }
    }
#pragma unroll
    for (int nt = 0; nt < 4; ++nt) {
        const int n = nt * 16 + l15;
#pragma unroll
        for (int r = 0; r < 8; ++r)
            h[(size_t)(row0 + r + half * 8) * DIM_E + n] = acc[nt][r];
    }
}

// ---------------------------------------------------------------------------
// K6: degree of each node (atomic count over edge rows) + rsqrt in place
// ---------------------------------------------------------------------------
__global__ void lattice_deg_kernel(const int* __restrict__ ei,
                                   float* __restrict__ deg) {
    const int e = blockIdx.x * blockDim.x + threadIdx.x;
    if (e < N_EDGE) atomicAdd(&deg[ei[e]], 1.0f);
}

__global__ void lattice_dinv_kernel(float* __restrict__ deg) {
    const int i = blockIdx.x * blockDim.x + threadIdx.x;
    if (i < NUM_NODES) deg[i] = rsqrtf(deg[i]);
}

// ---------------------------------------------------------------------------
// K7: one gcn_conv layer (wave per edge; 2 dims/lane; atomic scatter)
// ---------------------------------------------------------------------------
__global__ void lattice_gcn_kernel(const int* __restrict__ ei,
                                   const float* __restrict__ dinv,
                                   const float* __restrict__ x,
                                   float* __restrict__ out) {
    const int e = blockIdx.x * (blockDim.x >> 5) + (threadIdx.x >> 5);
    if (e >= N_EDGE) return;
    const int lane = threadIdx.x & 31;
    const int r = ei[e];
    const int c = ei[N_EDGE + e];
    const float nrm = dinv[r] * dinv[c];
    const float m0 = nrm * x[(size_t)r * DIM_E + lane];
    const float m1 = nrm * x[(size_t)r * DIM_E + 32 + lane];
    atomicAdd(&out[(size_t)c * DIM_E + lane], m0);
    atomicAdd(&out[(size_t)c * DIM_E + 32 + lane], m1);
}

// ---------------------------------------------------------------------------
// K8: out = (ego + x1 + x2)/3 ; item rows += h / ||h||
// ---------------------------------------------------------------------------
__global__ void lattice_final_kernel(const float* __restrict__ ego,
                                     const float* __restrict__ x1,
                                     const float* __restrict__ x2,
                                     const float* __restrict__ h,
                                     float* __restrict__ out) {
    const int row = blockIdx.x;
    const int lane = threadIdx.x;
    const size_t o = (size_t)row * DIM_E;
    float a0 = (ego[o + lane] + x1[o + lane] + x2[o + lane]) * (1.0f / 3.0f);
    float a1 = (ego[o + 32 + lane] + x1[o + 32 + lane] + x2[o + 32 + lane]) * (1.0f / 3.0f);
    if (row >= NUM_USER) {
        const size_t hi = (size_t)(row - NUM_USER) * DIM_E;
        const float h0 = h[hi + lane];
        const float h1 = h[hi + 32 + lane];
        float ss = h0 * h0 + h1 * h1;
#pragma unroll
        for (int off = 16; off > 0; off >>= 1) ss += __shfl_xor(ss, off, 32);
        const float rn = rsqrtf(ss);
        a0 += h0 * rn;
        a1 += h1 * rn;
    }
    out[o + lane] = a0;
    out[o + 32 + lane] = a1;
}

// ---------------------------------------------------------------------------
extern "C" void kernel_launch(void* const* d_in, const int* in_sizes, int n_in,
                              void* d_out, int out_size, void* d_ws, size_t ws_size,
                              hipStream_t stream) {
    const float* user_emb = (const float*)d_in[0];
    const float* item_emb = (const float*)d_in[1];
    const float* v_feat   = (const float*)d_in[2];
    const float* t_feat   = (const float*)d_in[3];
    const float* Wi       = (const float*)d_in[4];
    const float* bi       = (const float*)d_in[5];
    const float* Wt       = (const float*)d_in[6];
    const float* bt       = (const float*)d_in[7];
    const float* mw       = (const float*)d_in[8];
    const float* img_adj  = (const float*)d_in[9];
    const float* txt_adj  = (const float*)d_in[10];
    const int*   ei       = (const int*)d_in[11];
    float* out = (float*)d_out;

    // workspace layout (element offsets)
    float* ws = (float*)d_ws;
    float* f_tmp  = ws;                                  // NUM_ITEM*64
    float* f_n    = f_tmp  + (size_t)NUM_ITEM * DIM_E;   // NUM_ITEM*64
    float* tv_img = f_n    + (size_t)NUM_ITEM * DIM_E;   // NUM_ITEM*TOPK
    float* tv_txt = tv_img + (size_t)NUM_ITEM * TOPK;
    int*   ti_img = (int*)(tv_txt + (size_t)NUM_ITEM * TOPK);
    int*   ti_txt = ti_img + (size_t)NUM_ITEM * TOPK;
    float* dvec   = (float*)(ti_txt + (size_t)NUM_ITEM * TOPK); // NUM_ITEM
    float* h      = dvec + NUM_ITEM;                     // NUM_ITEM*64
    float* ego    = h    + (size_t)NUM_ITEM * DIM_E;     // NUM_NODES*64
    float* x1     = ego  + (size_t)NUM_NODES * DIM_E;
    float* x2     = x1   + (size_t)NUM_NODES * DIM_E;
    float* deg    = x2   + (size_t)NUM_NODES * DIM_E;    // NUM_NODES

    const int rowblocks = NUM_ITEM / 16;  // 384

    // --- image modality: project -> normalize -> sim + top-k ---------------
    lattice_proj_kernel<<<rowblocks, 32, 0, stream>>>(v_feat, Wi, bi, f_tmp, V_DIM);
    lattice_rownorm_kernel<<<NUM_ITEM, 32, 0, stream>>>(f_tmp, f_n);
    lattice_simtopk_kernel<<<rowblocks, 32, 0, stream>>>(f_n, tv_img, ti_img);

    // --- text modality ------------------------------------------------------
    lattice_proj_kernel<<<rowblocks, 32, 0, stream>>>(t_feat, Wt, bt, f_tmp, T_DIM);
    lattice_rownorm_kernel<<<NUM_ITEM, 32, 0, stream>>>(f_tmp, f_n);
    lattice_simtopk_kernel<<<rowblocks, 32, 0, stream>>>(f_n, tv_txt, ti_txt);

    // --- learned adjacency normalization + h propagation --------------------
    lattice_learned_d_kernel<<<(NUM_ITEM + 255) / 256, 256, 0, stream>>>(tv_img, tv_txt, mw, dvec);
    lattice_h_sparse_kernel<<<NUM_ITEM, 32, 0, stream>>>(tv_img, ti_img, tv_txt, ti_txt,
                                                         dvec, mw, item_emb, h);
    lattice_h_dense_kernel<<<rowblocks, 32, 0, stream>>>(img_adj, txt_adj, mw, item_emb, h);

    // --- LightGCN ------------------------------------------------------------
    hipMemcpyAsync(ego, user_emb, (size_t)NUM_USER * DIM_E * sizeof(float),
                   hipMemcpyDeviceToDevice, stream);
    hipMemcpyAsync(ego + (size_t)NUM_USER * DIM_E, item_emb,
                   (size_t)NUM_ITEM * DIM_E * sizeof(float),
                   hipMemcpyDeviceToDevice, stream);
    hipMemsetAsync(deg, 0, (size_t)NUM_NODES * sizeof(float), stream);
    lattice_deg_kernel<<<(N_EDGE + 255) / 256, 256, 0, stream>>>(ei, deg);
    lattice_dinv_kernel<<<(NUM_NODES + 255) / 256, 256, 0, stream>>>(deg);

    hipMemsetAsync(x1, 0, (size_t)NUM_NODES * DIM_E * sizeof(float), stream);
    lattice_gcn_kernel<<<(N_EDGE + 7) / 8, 256, 0, stream>>>(ei, deg, ego, x1);
    hipMemsetAsync(x2, 0, (size_t)NUM_NODES * DIM_E * sizeof(float), stream);
    lattice_gcn_kernel<<<(N_EDGE + 7) / 8, 256, 0, stream>>>(ei, deg, x1, x2);

    // --- finalize ------------------------------------------------------------
    lattice_final_kernel<<<NUM_NODES, 32, 0, stream>>>(ego, x1, x2, h, out);
}